// SelfOrganizingGraphEmbedding_85572928405571
// MI455X (gfx1250) — compile-verified
//
#include <hip/hip_runtime.h>
#include <hip/hip_bf16.h>

#define N_NODES 50000
#define N_EDGES 200000
#define NODE_DIM 128
#define EDGE_DIM 64
#define HIDDEN 128
#define N_LAYERS 3
#define LN_EPS 1e-6f

typedef __attribute__((ext_vector_type(2))) float v2f;
typedef __attribute__((ext_vector_type(8))) float v8f;

#if __has_builtin(__builtin_amdgcn_wmma_f32_16x16x4_f32)
#define HAVE_WMMA_F32 1
#else
#warning "wmma_f32_16x16x4_f32 builtin not available on this toolchain"
#define HAVE_WMMA_F32 0
#endif

static __device__ __forceinline__ v8f wmma4(v2f a, v2f b, v8f c) {
#if HAVE_WMMA_F32
  // 8 args: (neg_a, A, neg_b, B, c_mod, C, reuse_a, reuse_b)
  return __builtin_amdgcn_wmma_f32_16x16x4_f32(false, a, false, b, (short)0, c,
                                               false, false);
#else
  // placeholder so the file still compiles if the builtin is missing
  c[0] += a.x * b.x;
  return c;
#endif
}

// ---------------------------------------------------------------------------
// C[M,128] = A[M,K] @ W[K,128] + bias   (one 16x16 tile per wave, fp32 WMMA)
// ---------------------------------------------------------------------------
__global__ void sog_embed_gemm(const float* __restrict__ A,
                               const float* __restrict__ W,
                               const float* __restrict__ bias,
                               float* __restrict__ C, int M, int K) {
  const int lane = threadIdx.x & 31;
  const int wave = threadIdx.x >> 5;
  const int tile = blockIdx.x * 4 + wave;
  const int nTiles = (M >> 4) * (HIDDEN >> 4);
  if (tile >= nTiles) return;

  const int row0 = (tile >> 3) << 4;  // HIDDEN/16 == 8 column tiles
  const int col0 = (tile & 7) << 4;
  const int m = lane & 15;
  const int koff = (lane < 16) ? 0 : 2;
  const int n = col0 + (lane & 15);

  v8f acc = {0.f, 0.f, 0.f, 0.f, 0.f, 0.f, 0.f, 0.f};
  const float* arow = A + (size_t)(row0 + m) * K + koff;
  for (int k = 0; k < K; k += 4) {
    v2f a;
    a.x = arow[k];
    a.y = arow[k + 1];
    const float* wk = W + (size_t)(k + koff) * HIDDEN + n;
    v2f b;
    b.x = wk[0];
    b.y = wk[HIDDEN];
    acc = wmma4(a, b, acc);
  }

  const float bv = bias[n];
  const int half8 = (lane < 16) ? 0 : 8;
#pragma unroll
  for (int v = 0; v < 8; ++v)
    C[(size_t)(row0 + v + half8) * HIDDEN + n] = acc[v] + bv;
}

// ---------------------------------------------------------------------------
// zero fill (float4 per thread)
// ---------------------------------------------------------------------------
__global__ void sog_zero(float4* __restrict__ p, int n4) {
  int i = blockIdx.x * blockDim.x + threadIdx.x;
  if (i < n4) p[i] = make_float4(0.f, 0.f, 0.f, 0.f);
}

// ---------------------------------------------------------------------------
// Edge phase: one wave per edge. Gather src/dst rows, score with W_top,
// sigmoid, message = src*w + edge_h, scatter-add into agg via fp32 atomics.
// ---------------------------------------------------------------------------
__global__ void sog_edge(const float* __restrict__ node_h,
                         const float* __restrict__ edge_h,
                         const int* __restrict__ src,
                         const int* __restrict__ dst,
                         const float* __restrict__ W_top,
                         const float* __restrict__ b_top,
                         float* __restrict__ agg,
                         float* __restrict__ ew_out, int writeW) {
  const int lane = threadIdx.x & 31;
  const int wave = threadIdx.x >> 5;
  const int e = blockIdx.x * 8 + wave;
  if (e >= N_EDGES) return;

  const int s = src[e];
  const int d = dst[e];

  const float4 sv = ((const float4*)(node_h + (size_t)s * HIDDEN))[lane];
  const float4 dv = ((const float4*)(node_h + (size_t)d * HIDDEN))[lane];
  const float4 ev = ((const float4*)(edge_h + (size_t)e * HIDDEN))[lane];
  const float4 w1 = ((const float4*)(W_top + 0))[lane];
  const float4 w2 = ((const float4*)(W_top + HIDDEN))[lane];
  const float4 w3 = ((const float4*)(W_top + 2 * HIDDEN))[lane];

  float p = sv.x * w1.x + sv.y * w1.y + sv.z * w1.z + sv.w * w1.w +
            dv.x * w2.x + dv.y * w2.y + dv.z * w2.z + dv.w * w2.w +
            ev.x * w3.x + ev.y * w3.y + ev.z * w3.z + ev.w * w3.w;
#pragma unroll
  for (int mask = 16; mask >= 1; mask >>= 1) p += __shfl_xor(p, mask, 32);

  const float w = 1.f / (1.f + expf(-(p + b_top[0])));

  float* dp = agg + (size_t)d * HIDDEN + lane * 4;
#if defined(__HIP_DEVICE_COMPILE__)
  unsafeAtomicAdd(dp + 0, sv.x * w + ev.x);
  unsafeAtomicAdd(dp + 1, sv.y * w + ev.y);
  unsafeAtomicAdd(dp + 2, sv.z * w + ev.z);
  unsafeAtomicAdd(dp + 3, sv.w * w + ev.w);
#endif
  if (writeW && lane == 0) ew_out[e] = w;
}

// ---------------------------------------------------------------------------
// Node update: per wave, 16 rows x 128 cols.
//   upd = [node_h, agg] @ W_gnn + b_gnn ; x = node_h + upd
//   node_h = relu(LayerNorm(x))   (in place; row m depends only on row m)
// D layout: acc[t][v] holds row (row0+v+half8), col (t*16 + lane%16).
// ---------------------------------------------------------------------------
__global__ void sog_node_update(float* __restrict__ node_h,
                                const float* __restrict__ agg,
                                const float* __restrict__ Wg,
                                const float* __restrict__ bg,
                                const float* __restrict__ lns,
                                const float* __restrict__ lnb) {
  const int lane = threadIdx.x & 31;
  const int wave = threadIdx.x >> 5;
  const int rowTile = blockIdx.x * 4 + wave;
  if (rowTile >= (N_NODES >> 4)) return;

  const int row0 = rowTile << 4;
  const int m = lane & 15;
  const int koff = (lane < 16) ? 0 : 2;
  const int ncol = lane & 15;

  v8f acc[8];
#pragma unroll
  for (int t = 0; t < 8; ++t)
    acc[t] = (v8f){0.f, 0.f, 0.f, 0.f, 0.f, 0.f, 0.f, 0.f};

  // first half of K: A = node_h
  {
    const float* arow = node_h + (size_t)(row0 + m) * HIDDEN + koff;
    for (int k = 0; k < HIDDEN; k += 4) {
      v2f a;
      a.x = arow[k];
      a.y = arow[k + 1];
      const float* wk = Wg + (size_t)(k + koff) * HIDDEN;
#pragma unroll
      for (int t = 0; t < 8; ++t) {
        v2f b;
        b.x = wk[t * 16 + ncol];
        b.y = wk[HIDDEN + t * 16 + ncol];
        acc[t] = wmma4(a, b, acc[t]);
      }
    }
  }
  // second half of K: A = aggregated
  {
    const float* arow = agg + (size_t)(row0 + m) * HIDDEN + koff;
    for (int k = 0; k < HIDDEN; k += 4) {
      v2f a;
      a.x = arow[k];
      a.y = arow[k + 1];
      const float* wk = Wg + (size_t)(HIDDEN + k + koff) * HIDDEN;
#pragma unroll
      for (int t = 0; t < 8; ++t) {
        v2f b;
        b.x = wk[t * 16 + ncol];
        b.y = wk[HIDDEN + t * 16 + ncol];
        acc[t] = wmma4(a, b, acc[t]);
      }
    }
  }

  const int half8 = (lane < 16) ? 0 : 8;

  // residual + bias
#pragma unroll
  for (int t = 0; t < 8; ++t) {
    const int n = t * 16 + ncol;
    const float bb = bg[n];
#pragma unroll
    for (int v = 0; v < 8; ++v)
      acc[t][v] += bb + node_h[(size_t)(row0 + v + half8) * HIDDEN + n];
  }

  // LayerNorm over the 128 columns of each row (half-wave xor reduction)
  float mean[8], inv[8];
#pragma unroll
  for (int v = 0; v < 8; ++v) {
    float s = 0.f, q = 0.f;
#pragma unroll
    for (int t = 0; t < 8; ++t) {
      const float f = acc[t][v];
      s += f;
      q += f * f;
    }
#pragma unroll
    for (int mask = 8; mask >= 1; mask >>= 1) {
      s += __shfl_xor(s, mask, 32);
      q += __shfl_xor(q, mask, 32);
    }
    const float mu = s * (1.f / HIDDEN);
    mean[v] = mu;
    inv[v] = rsqrtf(q * (1.f / HIDDEN) - mu * mu + LN_EPS);
  }

  // scale/shift, relu, store in place
#pragma unroll
  for (int t = 0; t < 8; ++t) {
    const int n = t * 16 + ncol;
    const float sc = lns[n];
    const float sh = lnb[n];
#pragma unroll
    for (int v = 0; v < 8; ++v) {
      float y = (acc[t][v] - mean[v]) * inv[v] * sc + sh;
      y = fmaxf(y, 0.f);
      node_h[(size_t)(row0 + v + half8) * HIDDEN + n] = y;
    }
  }
}

// ---------------------------------------------------------------------------
extern "C" void kernel_launch(void* const* d_in, const int* in_sizes, int n_in,
                              void* d_out, int out_size, void* d_ws,
                              size_t ws_size, hipStream_t stream) {
  const float* node_features = (const float*)d_in[0];
  const float* edge_features = (const float*)d_in[1];
  const int* edge_indices = (const int*)d_in[2];
  const float* W_node = (const float*)d_in[3];
  const float* b_node = (const float*)d_in[4];
  const float* W_edge = (const float*)d_in[5];
  const float* b_edge = (const float*)d_in[6];
  const float* W_gnn = (const float*)d_in[7];
  const float* b_gnn = (const float*)d_in[8];
  const float* W_top = (const float*)d_in[9];
  const float* b_top = (const float*)d_in[10];
  const float* ln_scale = (const float*)d_in[11];
  const float* ln_bias = (const float*)d_in[12];

  float* node_h = (float*)d_out;                         // [N_NODES, HIDDEN]
  float* ew_out = node_h + (size_t)N_NODES * HIDDEN;     // [N_EDGES]
  float* edge_h = (float*)d_ws;                          // [N_EDGES, HIDDEN]
  float* agg = edge_h + (size_t)N_EDGES * HIDDEN;        // [N_NODES, HIDDEN]

  const int* src = edge_indices;
  const int* dst = edge_indices + N_EDGES;

  // embeddings
  {
    const int tiles = (N_NODES / 16) * (HIDDEN / 16);  // 25000
    sog_embed_gemm<<<tiles / 4, 128, 0, stream>>>(node_features, W_node, b_node,
                                                  node_h, N_NODES, NODE_DIM);
  }
  {
    const int tiles = (N_EDGES / 16) * (HIDDEN / 16);  // 100000
    sog_embed_gemm<<<tiles / 4, 128, 0, stream>>>(edge_features, W_edge, b_edge,
                                                  edge_h, N_EDGES, EDGE_DIM);
  }

  const int aggN4 = (N_NODES * HIDDEN) / 4;  // 1.6M float4
  for (int layer = 0; layer < N_LAYERS; ++layer) {
    sog_zero<<<(aggN4 + 255) / 256, 256, 0, stream>>>((float4*)agg, aggN4);
    sog_edge<<<N_EDGES / 8, 256, 0, stream>>>(node_h, edge_h, src, dst, W_top,
                                              b_top, agg, ew_out,
                                              layer == N_LAYERS - 1 ? 1 : 0);
    const int rowTiles = N_NODES / 16;  // 3125
    sog_node_update<<<(rowTiles + 3) / 4, 128, 0, stream>>>(
        node_h, agg, W_gnn + (size_t)layer * 2 * HIDDEN * HIDDEN,
        b_gnn + (size_t)layer * HIDDEN, ln_scale + (size_t)layer * HIDDEN,
        ln_bias + (size_t)layer * HIDDEN);
  }
}